// cola_gnn_84834194031069
// MI455X (gfx1250) — compile-verified
//
#include <hip/hip_runtime.h>
#include <hip/hip_bf16.h>
#include <math.h>

// Model dims
#define BB 8
#define WW 32
#define MM 512
#define HH 128
#define HF 64
#define KK 16
#define SP 64

typedef __attribute__((ext_vector_type(2))) float v2f;
typedef __attribute__((ext_vector_type(8))) float v8f;

// D = A(16x4) x B(4x16) + C, fp32, one WMMA per call (compiler inserts hazard NOPs)
__device__ __forceinline__ v8f wmma4(v2f a, v2f b, v8f c) {
  return __builtin_amdgcn_wmma_f32_16x16x4_f32(false, a, false, b, (short)0, c, false, false);
}

// ---------------------------------------------------------------------------
// 0) one-time transpose of Whh so the RNN B-matrix reads are coalesced:
//    WhhT[k*128 + n] = Whh[n*128 + k]
// ---------------------------------------------------------------------------
__global__ __launch_bounds__(256) void transpose_whh_kernel(const float* __restrict__ Whh,
                                                            float* __restrict__ WhhT) {
  int i = blockIdx.x * 256 + threadIdx.x;   // 128*128 = 16384
  if (i < HH * HH) {
    int k = i >> 7, n = i & 127;
    WhhT[k * HH + n] = Whh[n * HH + k];
  }
}

// ---------------------------------------------------------------------------
// 1) Temporal RNN: h = tanh(x_t*Wih + bih + h@Whh^T + bhh), 32 steps.
//    1 block = 16 sequences x 128 hidden; 8 waves, wave o owns cols [16o,16o+16).
// ---------------------------------------------------------------------------
__global__ __launch_bounds__(256) void rnn_kernel(const float* __restrict__ x,
                                                  const float* __restrict__ Wih,
                                                  const float* __restrict__ WhhT,
                                                  const float* __restrict__ bih,
                                                  const float* __restrict__ bhh,
                                                  float* __restrict__ last_hid) {
  __shared__ float sh[16 * 132];                 // h tile, padded stride 132
  const int tid  = threadIdx.x;
  const int wave = tid >> 5, lane = tid & 31;
  const int half = lane >> 4, l16 = lane & 15;
  const int tile = blockIdx.x;                   // 256 tiles of 16 rows
  const int b    = tile >> 5;                    // 32 tiles per batch
  const int m0   = (tile & 31) << 4;
  for (int i = tid; i < 16 * HH; i += 256) sh[(i >> 7) * 132 + (i & 127)] = 0.0f;
  __syncthreads();
  const int   n0     = wave << 4;
  const float wih_c  = Wih[n0 + l16];            // Wih is (H,1)
  const float bsum_c = bih[n0 + l16] + bhh[n0 + l16];
  for (int w = 0; w < WW; ++w) {
    v8f acc = {};
    for (int kc = 0; kc < HH; kc += 4) {
      v2f av, bm;
      av.x = sh[l16 * 132 + kc + 2 * half];
      av.y = sh[l16 * 132 + kc + 2 * half + 1];
      bm.x = WhhT[(kc + 2 * half) * HH + n0 + l16];      // B[k][n] = Whh[n][k]
      bm.y = WhhT[(kc + 2 * half + 1) * HH + n0 + l16];
      acc  = wmma4(av, bm, acc);
    }
    __syncthreads();   // all waves done reading sh for this step
#pragma unroll
    for (int v = 0; v < 8; ++v) {
      int   row = v + 8 * half;
      float xv  = x[(size_t)b * (WW * MM) + (size_t)w * MM + m0 + row];
      sh[row * 132 + n0 + l16] = tanhf(acc[v] + xv * wih_c + bsum_c);
    }
    __syncthreads();
  }
  for (int i = tid; i < 16 * HH; i += 256) {
    int r = i >> 7, c = i & 127;
    last_hid[((size_t)b * MM + m0 + r) * HH + c] = sh[r * 132 + c];
  }
}

// ---------------------------------------------------------------------------
// 2) s1 = lh@W1^T ; s2 = lh@W2^T + b1  (waves 0-3 -> s1, waves 4-7 -> s2)
// ---------------------------------------------------------------------------
__global__ __launch_bounds__(256) void proj_kernel(const float* __restrict__ lh,
                                                   const float* __restrict__ W1,
                                                   const float* __restrict__ W2,
                                                   const float* __restrict__ b1,
                                                   float* __restrict__ s1,
                                                   float* __restrict__ s2) {
  __shared__ float sA[16 * 132];
  const int tid = threadIdx.x, wave = tid >> 5, lane = tid & 31;
  const int half = lane >> 4, l16 = lane & 15;
  const int row0 = blockIdx.x << 4;
  for (int i = tid; i < 16 * HH; i += 256)
    sA[(i >> 7) * 132 + (i & 127)] = lh[(size_t)row0 * HH + i];
  __syncthreads();
  const int    which = wave >> 2;
  const int    n0    = (wave & 3) << 4;
  const float* Wm    = which ? W2 : W1;
  v8f acc = {};
  for (int kc = 0; kc < HH; kc += 4) {
    v2f av, bm;
    av.x = sA[l16 * 132 + kc + 2 * half];
    av.y = sA[l16 * 132 + kc + 2 * half + 1];
    bm.x = Wm[(n0 + l16) * HH + kc + 2 * half];          // B[k][n] = W[n][k]
    bm.y = Wm[(n0 + l16) * HH + kc + 2 * half + 1];
    acc  = wmma4(av, bm, acc);
  }
  const float addb = which ? b1[n0 + l16] : 0.0f;
  float*      outp = which ? s2 : s1;
#pragma unroll
  for (int v = 0; v < 8; ++v) {
    int row = v + 8 * half;
    outp[(size_t)(row0 + row) * HF + n0 + l16] = acc[v] + addb;
  }
}

// ---------------------------------------------------------------------------
// 3) a[b,i,j] = V . elu(s1[b,j,:] + s2b1[b,i,:]) + bv   (1 block per (b,i))
// ---------------------------------------------------------------------------
__global__ __launch_bounds__(256) void attn_a_kernel(const float* __restrict__ s1,
                                                     const float* __restrict__ s2,
                                                     const float* __restrict__ V,
                                                     const float* __restrict__ bv,
                                                     float* __restrict__ a) {
  __shared__ float srow[HF], sV[HF];
  const int bi = blockIdx.x;        // b*512 + i
  const int b  = bi >> 9;
  const int tid = threadIdx.x;
  if (tid < HF) { srow[tid] = s2[(size_t)bi * HF + tid]; sV[tid] = V[tid]; }
  __syncthreads();
  const float bv0 = bv[0];
  for (int j = tid; j < MM; j += 256) {
    const float* p = s1 + ((size_t)b * MM + j) * HF;
    float sum = 0.0f;
#pragma unroll 8
    for (int h = 0; h < HF; ++h) {
      float t = p[h] + srow[h];
      float e = t > 0.0f ? t : expm1f(t);
      sum += e * sV[h];
    }
    a[(size_t)bi * MM + j] = sum + bv0;
  }
}

// 4) inv-norm over axis i (columns j coalesced)
__global__ __launch_bounds__(256) void colnorm_kernel(const float* __restrict__ a,
                                                      float* __restrict__ inv) {
  const int b = blockIdx.x >> 1;
  const int j = ((blockIdx.x & 1) << 8) + threadIdx.x;
  const float* ab = a + (size_t)b * MM * MM;
  float s = 0.0f;
  for (int i = 0; i < MM; ++i) { float v = ab[(size_t)i * MM + j]; s += v * v; }
  inv[b * MM + j] = 1.0f / fmaxf(sqrtf(s), 1e-12f);
}

__global__ __launch_bounds__(256) void colscale_kernel(float* __restrict__ a,
                                                       const float* __restrict__ inv) {
  size_t idx = (size_t)blockIdx.x * 256 + threadIdx.x;   // 8*512*512
  int    j   = idx & (MM - 1);
  int    b   = (int)(idx >> 18);
  a[idx] *= inv[b * MM + j];
}

// ---------------------------------------------------------------------------
// 5) conv features: feat[b,m,3k+{0,1,2}] = relu(r, rl0, rl1)
// ---------------------------------------------------------------------------
__global__ __launch_bounds__(256) void conv_kernel(const float* __restrict__ x,
                                                   const float* __restrict__ conv_w,
                                                   const float* __restrict__ conv_b,
                                                   const float* __restrict__ convl_w,
                                                   const float* __restrict__ convl_b,
                                                   float* __restrict__ feat) {
  __shared__ float scw[KK * WW], sclw[KK * (WW - KK)];
  const int tid = threadIdx.x;
  for (int i = tid; i < KK * WW; i += 256) scw[i] = conv_w[i];
  for (int i = tid; i < KK * (WW - KK); i += 256) sclw[i] = convl_w[i];
  __syncthreads();
  const int b = blockIdx.x >> 5;
  const int m = ((blockIdx.x & 31) << 4) + (tid & 15);
  const int k = tid >> 4;
  const float* xp = x + (size_t)b * (WW * MM) + m;
  float r = conv_b[k], rl0 = convl_b[k], rl1 = convl_b[k];
#pragma unroll
  for (int t = 0; t < WW; ++t) r += xp[(size_t)t * MM] * scw[k * WW + t];
#pragma unroll
  for (int t = 0; t < WW - KK; ++t) {
    rl0 += xp[(size_t)(2 * t) * MM] * sclw[k * (WW - KK) + t];
    rl1 += xp[(size_t)(2 * t + 1) * MM] * sclw[k * (WW - KK) + t];
  }
  float* fp = feat + ((size_t)b * MM + m) * (3 * KK) + 3 * k;
  fp[0] = fmaxf(r, 0.0f);
  fp[1] = fmaxf(rl0, 0.0f);
  fp[2] = fmaxf(rl1, 0.0f);
}

// ---------------------------------------------------------------------------
// 6) g1 = feat @ gc1_w   (4096x48)@(48x128), WMMA, K=48
// ---------------------------------------------------------------------------
__global__ __launch_bounds__(256) void featgc1_kernel(const float* __restrict__ feat,
                                                      const float* __restrict__ gc1_w,
                                                      float* __restrict__ g1) {
  __shared__ float sA[16 * 52];
  const int tid = threadIdx.x, wave = tid >> 5, lane = tid & 31;
  const int half = lane >> 4, l16 = lane & 15;
  const int row0 = blockIdx.x << 4;
  for (int i = tid; i < 16 * 48; i += 256)
    sA[(i / 48) * 52 + (i % 48)] = feat[(size_t)row0 * 48 + i];
  __syncthreads();
  const int n0 = wave << 4;
  v8f acc = {};
  for (int kc = 0; kc < 48; kc += 4) {
    v2f av, bm;
    av.x = sA[l16 * 52 + kc + 2 * half];
    av.y = sA[l16 * 52 + kc + 2 * half + 1];
    bm.x = gc1_w[(kc + 2 * half) * HH + n0 + l16];
    bm.y = gc1_w[(kc + 2 * half + 1) * HH + n0 + l16];
    acc  = wmma4(av, bm, acc);
  }
#pragma unroll
  for (int v = 0; v < 8; ++v) {
    int row = v + 8 * half;
    g1[(size_t)(row0 + row) * HH + n0 + l16] = acc[v];
  }
}

// ---------------------------------------------------------------------------
// 7) gate GEMM + mix: c = sigmoid(a@Wb + wb); Amix = adj*c + a*(1-c)
//    per block: 16 rows x 512 cols (4 groups of 8 wave-tiles), K=512
// ---------------------------------------------------------------------------
__global__ __launch_bounds__(256) void awb_mix_kernel(const float* __restrict__ a,
                                                      const float* __restrict__ Wb,
                                                      const float* __restrict__ wb,
                                                      const float* __restrict__ adj,
                                                      float* __restrict__ Amix) {
  __shared__ float sA[16 * 516];
  const int tid = threadIdx.x, wave = tid >> 5, lane = tid & 31;
  const int half = lane >> 4, l16 = lane & 15;
  const int bt = blockIdx.x;            // 8*32
  const int b  = bt >> 5;
  const int i0 = (bt & 31) << 4;
  const float* arow = a + ((size_t)b * MM + i0) * MM;
  for (int i = tid; i < 16 * MM; i += 256)
    sA[(i >> 9) * 516 + (i & 511)] = arow[i];
  __syncthreads();
  const float wb0 = wb[0];
  for (int g = 0; g < 4; ++g) {
    const int n0 = g * 128 + (wave << 4);
    v8f acc = {};
    for (int kc = 0; kc < MM; kc += 4) {
      v2f av, bm;
      av.x = sA[l16 * 516 + kc + 2 * half];
      av.y = sA[l16 * 516 + kc + 2 * half + 1];
      bm.x = Wb[(size_t)(kc + 2 * half) * MM + n0 + l16];
      bm.y = Wb[(size_t)(kc + 2 * half + 1) * MM + n0 + l16];
      acc  = wmma4(av, bm, acc);
    }
#pragma unroll
    for (int v = 0; v < 8; ++v) {
      int   row = v + 8 * half;
      int   col = n0 + l16;
      float cg  = 1.0f / (1.0f + __expf(-(acc[v] + wb0)));
      float av2 = sA[row * 516 + col];
      float mix = adj[(size_t)(i0 + row) * MM + col] * cg + av2 * (1.0f - cg);
      Amix[((size_t)b * MM + i0 + row) * MM + col] = mix;
    }
  }
}

// ---------------------------------------------------------------------------
// 8) out = relu(Amix @ Bm + bias)  — per-batch dense graph conv, K=512, N<=128
// ---------------------------------------------------------------------------
__global__ __launch_bounds__(256) void spmm_relu_kernel(const float* __restrict__ Amix,
                                                        const float* __restrict__ Bm,
                                                        const float* __restrict__ bias,
                                                        float* __restrict__ outp, int N) {
  __shared__ float sA[16 * 516];
  const int tid = threadIdx.x, wave = tid >> 5, lane = tid & 31;
  const int half = lane >> 4, l16 = lane & 15;
  const int bt = blockIdx.x;
  const int b  = bt >> 5;
  const int i0 = (bt & 31) << 4;
  const float* arow = Amix + ((size_t)b * MM + i0) * MM;
  for (int i = tid; i < 16 * MM; i += 256)
    sA[(i >> 9) * 516 + (i & 511)] = arow[i];
  __syncthreads();
  const int n0 = wave << 4;
  if (n0 >= N) return;                     // N=64: waves 4-7 done (after barrier)
  const float* Bb = Bm + (size_t)b * MM * N;
  v8f acc = {};
  for (int kc = 0; kc < MM; kc += 4) {
    v2f av, bm;
    av.x = sA[l16 * 516 + kc + 2 * half];
    av.y = sA[l16 * 516 + kc + 2 * half + 1];
    bm.x = Bb[(size_t)(kc + 2 * half) * N + n0 + l16];
    bm.y = Bb[(size_t)(kc + 2 * half + 1) * N + n0 + l16];
    acc  = wmma4(av, bm, acc);
  }
  const float bcol = bias[n0 + l16];
#pragma unroll
  for (int v = 0; v < 8; ++v) {
    int row = v + 8 * half;
    outp[((size_t)b * MM + i0 + row) * N + n0 + l16] = fmaxf(acc[v] + bcol, 0.0f);
  }
}

// ---------------------------------------------------------------------------
// 9) g2 = h1 @ gc2_w  (4096x128)@(128x64), 4 waves/block
// ---------------------------------------------------------------------------
__global__ __launch_bounds__(128) void h1gc2_kernel(const float* __restrict__ h1,
                                                    const float* __restrict__ gc2_w,
                                                    float* __restrict__ g2) {
  __shared__ float sA[16 * 132];
  const int tid = threadIdx.x, wave = tid >> 5, lane = tid & 31;
  const int half = lane >> 4, l16 = lane & 15;
  const int row0 = blockIdx.x << 4;
  for (int i = tid; i < 16 * HH; i += 128)
    sA[(i >> 7) * 132 + (i & 127)] = h1[(size_t)row0 * HH + i];
  __syncthreads();
  const int n0 = wave << 4;
  v8f acc = {};
  for (int kc = 0; kc < HH; kc += 4) {
    v2f av, bm;
    av.x = sA[l16 * 132 + kc + 2 * half];
    av.y = sA[l16 * 132 + kc + 2 * half + 1];
    bm.x = gc2_w[(kc + 2 * half) * SP + n0 + l16];
    bm.y = gc2_w[(kc + 2 * half + 1) * SP + n0 + l16];
    acc  = wmma4(av, bm, acc);
  }
#pragma unroll
  for (int v = 0; v < 8; ++v) {
    int row = v + 8 * half;
    g2[(size_t)(row0 + row) * SP + n0 + l16] = acc[v];
  }
}

// 10) output head
__global__ __launch_bounds__(256) void head_kernel(const float* __restrict__ spat,
                                                   const float* __restrict__ lh,
                                                   const float* __restrict__ out_w,
                                                   const float* __restrict__ out_b,
                                                   float* __restrict__ out) {
  const int idx = blockIdx.x * 256 + threadIdx.x;   // 4096
  const float* sp = spat + (size_t)idx * SP;
  const float* hp = lh + (size_t)idx * HH;
  float s = out_b[0];
#pragma unroll 8
  for (int f = 0; f < SP; ++f) s += sp[f] * out_w[f];
#pragma unroll 8
  for (int h = 0; h < HH; ++h) s += hp[h] * out_w[SP + h];
  out[idx] = s;
}

extern "C" void kernel_launch(void* const* d_in, const int* in_sizes, int n_in,
                              void* d_out, int out_size, void* d_ws, size_t ws_size,
                              hipStream_t stream) {
  const float* x       = (const float*)d_in[0];
  const float* adj     = (const float*)d_in[1];
  const float* Wih     = (const float*)d_in[2];
  const float* Whh     = (const float*)d_in[3];
  const float* bih     = (const float*)d_in[4];
  const float* bhh     = (const float*)d_in[5];
  const float* W1      = (const float*)d_in[6];
  const float* W2      = (const float*)d_in[7];
  const float* b1      = (const float*)d_in[8];
  const float* V       = (const float*)d_in[9];
  const float* bv      = (const float*)d_in[10];
  const float* Wb      = (const float*)d_in[11];
  const float* wb      = (const float*)d_in[12];
  const float* conv_w  = (const float*)d_in[13];
  const float* conv_b  = (const float*)d_in[14];
  const float* convl_w = (const float*)d_in[15];
  const float* convl_b = (const float*)d_in[16];
  const float* gc1_w   = (const float*)d_in[17];
  const float* gc1_b   = (const float*)d_in[18];
  const float* gc2_w   = (const float*)d_in[19];
  const float* gc2_b   = (const float*)d_in[20];
  const float* out_w   = (const float*)d_in[21];
  const float* out_b   = (const float*)d_in[22];

  float* ws = (float*)d_ws;
  float* whhT     = ws; ws += HH * HH;            // 16K
  float* last_hid = ws; ws += 4096 * HH;          // 512K
  float* s1       = ws; ws += 4096 * HF;          // 256K
  float* s2       = ws; ws += 4096 * HF;          // 256K
  float* amat     = ws; ws += (size_t)BB * MM * MM; // 2M
  float* invn     = ws; ws += BB * MM;            // 4K
  float* feat     = ws; ws += 4096 * 48;          // 192K
  float* g1       = ws; ws += 4096 * HH;          // 512K
  float* Amix     = ws; ws += (size_t)BB * MM * MM; // 2M
  float* h1       = ws; ws += 4096 * HH;          // 512K
  float* g2       = ws; ws += 4096 * SP;          // 256K
  float* spat     = ws; ws += 4096 * SP;          // 256K

  transpose_whh_kernel<<<64, 256, 0, stream>>>(Whh, whhT);
  rnn_kernel<<<256, 256, 0, stream>>>(x, Wih, whhT, bih, bhh, last_hid);
  proj_kernel<<<256, 256, 0, stream>>>(last_hid, W1, W2, b1, s1, s2);
  attn_a_kernel<<<4096, 256, 0, stream>>>(s1, s2, V, bv, amat);
  colnorm_kernel<<<16, 256, 0, stream>>>(amat, invn);
  colscale_kernel<<<8192, 256, 0, stream>>>(amat, invn);
  conv_kernel<<<256, 256, 0, stream>>>(x, conv_w, conv_b, convl_w, convl_b, feat);
  featgc1_kernel<<<256, 256, 0, stream>>>(feat, gc1_w, g1);
  awb_mix_kernel<<<256, 256, 0, stream>>>(amat, Wb, wb, adj, Amix);
  spmm_relu_kernel<<<256, 256, 0, stream>>>(Amix, g1, gc1_b, h1, HH);
  h1gc2_kernel<<<256, 128, 0, stream>>>(h1, gc2_w, g2);
  spmm_relu_kernel<<<256, 256, 0, stream>>>(Amix, g2, gc2_b, spat, SP);
  head_kernel<<<16, 256, 0, stream>>>(spat, last_hid, out_w, out_b, (float*)d_out);
}